// ViTWarpV8_29360396435700
// MI455X (gfx1250) — compile-verified
//
#include <hip/hip_runtime.h>

// Problem shape (fixed by the reference).
#define BB 4
#define CC 64
#define HH 384
#define WW 768
#define HW (HH * WW)            // 294912
#define TP 256                  // pixels per block tile
#define WAVE_CH 8               // channels loaded per wave's TDM descriptor (64 ch / 8 waves)

typedef __attribute__((ext_vector_type(4))) unsigned int v4u;
typedef __attribute__((ext_vector_type(8))) int          v8i;
typedef __attribute__((ext_vector_type(4))) int          v4i;

// ---------------------------------------------------------------------------
// Tensor Data Mover: 2-D tile load (rows = channels with stride `stride0`
// elements, cols = contiguous pixels) from global into LDS.
// D# layout per CDNA5 ISA ch.8 (group0 128b + group1 256b; trailing groups
// zeroed — tile_dim2..4 == 0 so higher dims are unused).
// ---------------------------------------------------------------------------
__device__ __forceinline__ void tdm_load_2d(unsigned long long gaddr,   // byte address of tile start
                                            unsigned lds_off,          // LDS byte offset
                                            unsigned tdim0,            // tensor dim0 (elements)
                                            unsigned tdim1,            // tensor dim1 (elements)
                                            unsigned tile0,            // tile dim0 (elements, contiguous)
                                            unsigned tile1,            // tile dim1 (rows)
                                            unsigned long long stride0)// row stride (elements)
{
    v4u g0;
    g0[0] = 1u;                                               // count=1 valid descriptor
    g0[1] = lds_off;                                          // lds_addr
    g0[2] = (unsigned)(gaddr & 0xFFFFFFFFull);                // global_addr[31:0]
    g0[3] = (unsigned)((gaddr >> 32) & 0x1FFFFFFull)          // global_addr[56:32]
          | (2u << 30);                                       // type = 2 ("image")
    v8i g1;
    g1[0] = (int)(2u << 16);                                  // data_size = 2 (4 bytes)
    g1[1] = (int)((tdim0 & 0xFFFFu) << 16);                   // tensor_dim0[15:0]
    g1[2] = (int)(((tdim0 >> 16) & 0xFFFFu)                   // tensor_dim0[31:16]
          | ((tdim1 & 0xFFFFu) << 16));                       // tensor_dim1[15:0]
    g1[3] = (int)(((tdim1 >> 16) & 0xFFFFu)                   // tensor_dim1[31:16]
          | ((tile0 & 0xFFFFu) << 16));                       // tile_dim0
    g1[4] = (int)(tile1 & 0xFFFFu);                           // tile_dim1 (tile_dim2 = 0)
    g1[5] = (int)(stride0 & 0xFFFFFFFFull);                   // tensor_dim0_stride[31:0]
    g1[6] = (int)((stride0 >> 32) & 0xFFFFull);               // stride[47:32] (dim1 stride unused)
    g1[7] = 0;
    v4i z4 = {0, 0, 0, 0};
    v8i z8 = {0, 0, 0, 0, 0, 0, 0, 0};
    // clang-23 / therock-10.0 arity: (g0, g1, g2, g3, g4, cpol)
    __builtin_amdgcn_tensor_load_to_lds(g0, g1, z4, z4, z8, /*cpol=*/0);
}

// ---------------------------------------------------------------------------
// Kernel 1: clear accumulators (feature planes + weight plane) with b128 stores
// ---------------------------------------------------------------------------
__global__ __launch_bounds__(256) void k_zero(float* __restrict__ out, long long n)
{
    long long i = ((long long)blockIdx.x * blockDim.x + threadIdx.x) * 4;
    const long long stride = (long long)gridDim.x * blockDim.x * 4;
    float4 z{0.f, 0.f, 0.f, 0.f};
    for (; i + 3 < n; i += stride)
        *(float4*)(out + i) = z;
    if (blockIdx.x == 0 && threadIdx.x == 0)
        for (long long j = n & ~3ll; j < n; ++j) out[j] = 0.f;
}

// ---------------------------------------------------------------------------
// Kernel 2: scatter-splat. One block = 256 contiguous source pixels of one
// batch. Each wave TDM-loads its 8-channel x 256-pixel sub-tile into LDS,
// then every thread issues 1 weight atomic + 64 feature atomics.
// ---------------------------------------------------------------------------
__global__ __launch_bounds__(TP) void k_scatter(const float* __restrict__ fmap,
                                                const float* __restrict__ flow,
                                                const float* __restrict__ depth,
                                                float* __restrict__ out)
{
    __shared__ float tile[CC * TP];                 // 64 KB

    const int tilesPerB = HW / TP;                  // 1152
    const int b  = blockIdx.x / tilesPerB;
    const int t  = blockIdx.x - b * tilesPerB;
    const int p0 = t * TP;
    const int tid = threadIdx.x;

    // --- every wave issues its own TDM descriptor (EXEC is ignored by TDM,
    //     so we deliberately avoid divergent branching around it) ---
    const int wid = __builtin_amdgcn_readfirstlane((int)(threadIdx.x >> 5)); // wave id 0..7
    {
        const unsigned long long gsrc =
            (unsigned long long)(const void*)(fmap + ((size_t)b * CC + (size_t)wid * WAVE_CH) * HW + p0);
        const unsigned lds_off =
            (unsigned)(unsigned long long)(void*)tile + (unsigned)(wid * WAVE_CH * TP * 4);
        tdm_load_2d(gsrc, lds_off,
                    /*tdim0=*/HW, /*tdim1=*/WAVE_CH,
                    /*tile0=*/TP, /*tile1=*/WAVE_CH,
                    /*stride0=*/(unsigned long long)HW);
    }

    // --- overlap: per-pixel splat parameters while the TDM streams ---
    const int p  = p0 + tid;
    const float fx = flow[((size_t)b * 2)     * HW + p];
    const float fy = flow[((size_t)b * 2 + 1) * HW + p];
    const float dp = depth[(size_t)b * HW + p];
    const int gy = p / WW;
    const int gx = p - gy * WW;
    const float tx = (float)gx + fx;
    const float ty = (float)gy + fy;
    const bool valid = (tx >= 0.f) && (tx < (float)(WW - 1)) &&
                       (ty >= 0.f) && (ty < (float)(HH - 1));

    __builtin_amdgcn_s_wait_tensorcnt(0);           // wave's own TDM done
    __syncthreads();                                 // LDS tile visible block-wide

    if (valid) {
        const int ix  = (int)tx;                     // trunc == floor (tx >= 0)
        const int iy  = (int)ty;
        const int idx = iy * WW + ix;
        const float w = __expf(-dp);

        // weight accumulator lives in the mask plane of d_out
        atomicAdd(out + (size_t)BB * CC * HW + (size_t)b * HW + idx, w);

        float* fdst = out + (size_t)b * CC * HW + idx;
        #pragma unroll                                // full: 64 ds loads + 64 atomics straight-line
        for (int c = 0; c < CC; ++c) {
            atomicAdd(fdst + (size_t)c * HW, tile[c * TP + tid] * w);
        }
    }
}

// ---------------------------------------------------------------------------
// Kernel 3: in-place normalize + mask emit, 4 pixels per thread (b128 RMW).
// ---------------------------------------------------------------------------
__global__ __launch_bounds__(256) void k_norm(float* __restrict__ out)
{
    const int q = blockIdx.x * blockDim.x + threadIdx.x;   // quad id, 0 .. B*HW/4-1
    const int i = q * 4;                                   // first pixel of quad
    const int b = i / HW;
    const int p = i - b * HW;

    float4* wptr = (float4*)(out + (size_t)BB * CC * HW) + q;
    const float4 w = *wptr;
    float4 inv, msk;
    inv.x = w.x > 0.f ? 1.f / w.x : 1.f;  msk.x = w.x > 0.f ? 1.f : 0.f;
    inv.y = w.y > 0.f ? 1.f / w.y : 1.f;  msk.y = w.y > 0.f ? 1.f : 0.f;
    inv.z = w.z > 0.f ? 1.f / w.z : 1.f;  msk.z = w.z > 0.f ? 1.f : 0.f;
    inv.w = w.w > 0.f ? 1.f / w.w : 1.f;  msk.w = w.w > 0.f ? 1.f : 0.f;

    float* f = out + (size_t)b * CC * HW + p;
    #pragma unroll 16
    for (int c = 0; c < CC; ++c) {
        float4* fp = (float4*)(f + (size_t)c * HW);
        float4 v = *fp;
        v.x *= inv.x; v.y *= inv.y; v.z *= inv.z; v.w *= inv.w;
        *fp = v;
    }

    *wptr = msk;
}

// ---------------------------------------------------------------------------
extern "C" void kernel_launch(void* const* d_in, const int* in_sizes, int n_in,
                              void* d_out, int out_size, void* d_ws, size_t ws_size,
                              hipStream_t stream)
{
    const float* fmap  = (const float*)d_in[0];   // (B, C, H, W) f32
    const float* flow  = (const float*)d_in[1];   // (B, 2, H, W) f32
    const float* depth = (const float*)d_in[2];   // (B, 1, H, W) f32
    float* out = (float*)d_out;                   // features then mask, flat

    const long long n = (long long)BB * CC * HW + (long long)BB * HW; // == out_size

    k_zero<<<4096, 256, 0, stream>>>(out, n);

    const int nblk = BB * (HW / TP);              // 4608
    k_scatter<<<nblk, TP, 0, stream>>>(fmap, flow, depth, out);

    k_norm<<<(BB * HW / 4) / 256, 256, 0, stream>>>(out);
}